// TempoStateRNNCell_58806692217077
// MI455X (gfx1250) — compile-verified
//
#include <hip/hip_runtime.h>
#include <stdint.h>

// TempoStateRNNCell: c[t] = a[t]*c[t-1] + b[t] over t=0..2047, for 16x2048
// independent sequences. Bandwidth-bound (768 MB min traffic, ~33us at
// 23.3 TB/s). Block = 16 waves x 32 units; time axis split into 16 segments
// per block (affine composition is associative), carries combined in LDS.
// Loads are staged global->LDS with CDNA5 async-copy (ASYNCcnt) so each wave
// keeps its segment stream in flight without VGPR pressure. Output stores are
// non-temporal so the phase-A input lines survive in L2 for the phase-B replay.

namespace {

constexpr int    kUnits  = 2048;            // feature dim of a (and of b)
constexpr int    kSeq    = 2048;            // scan length
constexpr int    kBatch  = 16;
constexpr int    kWaves  = 16;              // waves per block (512 threads)
constexpr int    kSeg    = kSeq / kWaves;   // 128 timesteps per wave segment
constexpr int    kCT     = 16;              // timesteps staged per async chunk
constexpr int    kChunks = kSeg / kCT;      // 8 chunks per segment
constexpr size_t kRow    = 2 * kUnits;      // 4096 floats per (batch,t) row

#define HAS_ASYNC_LDS  __has_builtin(__builtin_amdgcn_global_load_async_to_lds_b128)
#define HAS_WAIT_ASYNC __has_builtin(__builtin_amdgcn_s_wait_asynccnt)

#if HAS_ASYNC_LDS
typedef int v4i __attribute__((ext_vector_type(4)));
typedef __attribute__((address_space(1))) v4i* as1_v4i_ptr;
typedef __attribute__((address_space(3))) v4i* as3_v4i_ptr;

// One lane moves 16B global -> 16B LDS asynchronously (GLOBAL_LOAD_ASYNC_TO_LDS_B128).
__device__ __forceinline__ void async_copy16(const float* gsrc, float* ldst) {
  __builtin_amdgcn_global_load_async_to_lds_b128((as1_v4i_ptr)gsrc, (as3_v4i_ptr)ldst,
                                                 /*offset=*/0, /*cpol=*/0);
}
#endif

__device__ __forceinline__ void wait_async_le8() {
#if HAS_WAIT_ASYNC
  __builtin_amdgcn_s_wait_asynccnt(8);
#elif HAS_ASYNC_LDS
  asm volatile("s_wait_asynccnt 8" ::: "memory");
#endif
}
__device__ __forceinline__ void wait_async_le0() {
#if HAS_WAIT_ASYNC
  __builtin_amdgcn_s_wait_asynccnt(0);
#elif HAS_ASYNC_LDS
  asm volatile("s_wait_asynccnt 0" ::: "memory");
#endif
}
__device__ __forceinline__ void wait_ds0() {
#if HAS_ASYNC_LDS
  asm volatile("s_wait_dscnt 0" ::: "memory");  // WAR guard before buffer reuse
#endif
}

__global__ __launch_bounds__(kWaves * 32)
void tempo_scan_kernel(const float* __restrict__ in, float* __restrict__ out) {
#if HAS_ASYNC_LDS
  // Per-wave double-buffered staging tiles: [wave][buf][t_local*32 + unit_lane]
  __shared__ float sTa[kWaves][2][kCT * 32];   // 64 KB
  __shared__ float sTb[kWaves][2][kCT * 32];   // 64 KB
#endif
  __shared__ float sA[kWaves][32];             // segment-composed A
  __shared__ float sB[kWaves][32];             // segment-composed B
  __shared__ float sC[kWaves][32];             // carry into each segment

  const int lane = threadIdx.x & 31;           // unit within group
  const int w    = threadIdx.x >> 5;           // wave = time segment
  const int ucol = blockIdx.x * 32;            // first unit column of group
  const int bt   = blockIdx.y;                 // batch
  const int t0   = w * kSeg;

  // in[bt][t][col]: base points at (bt, t=0, ucol); b columns are +kUnits.
  const float* base = in + (size_t)bt * kSeq * kRow + ucol;

  float Ac = 1.0f, Bc = 0.0f;                  // running affine of this segment

#if HAS_ASYNC_LDS
  // Async tile = kCT rows x 32 floats (128B/row). One B128 op per lane covers
  // 4 rows: lane -> row (lane>>3), 16B column chunk ((lane&7)*16B).
  const int r  = lane >> 3;
  const int c4 = (lane & 7) << 2;
  const float* gl = base + (size_t)(t0 + r) * kRow + c4;

  auto issue = [&](int tc, int buf) {
    const float* g = gl + (size_t)tc * kRow;
    float* la = &sTa[w][buf][r * 32 + c4];
    float* lb = &sTb[w][buf][r * 32 + c4];
#pragma unroll
    for (int i = 0; i < kCT / 4; ++i) {        // 4 sub-tiles of 4 rows each
      async_copy16(g + (size_t)(4 * i) * kRow,          la + 4 * i * 32);
      async_copy16(g + (size_t)(4 * i) * kRow + kUnits, lb + 4 * i * 32);
    }
  };

  // ---------------- Phase A: per-segment affine composition ----------------
  issue(0, 0);
  issue(kCT, 1);
#pragma unroll
  for (int ck = 0; ck < kChunks; ++ck) {
    if (ck + 1 < kChunks) wait_async_le8(); else wait_async_le0();
    const float* pa = &sTa[w][ck & 1][lane];
    const float* pb = &sTb[w][ck & 1][lane];
#pragma unroll
    for (int tl = 0; tl < kCT; ++tl) {
      const float la = pa[tl * 32];            // ds_load_b32, bank-conflict-free
      const float lb = pb[tl * 32];
      Ac = la * Ac;                            // A' = a*A
      Bc = fmaf(la, Bc, lb);                   // B' = a*B + b
    }
    if (ck + 2 < kChunks) { wait_ds0(); issue((ck + 2) * kCT, ck & 1); }
  }
#else
  {
    const float* g = base + (size_t)t0 * kRow + lane;
#pragma unroll 4
    for (int t = 0; t < kSeg; ++t) {
      const float la = g[(size_t)t * kRow];
      const float lb = g[(size_t)t * kRow + kUnits];
      Ac = la * Ac;
      Bc = fmaf(la, Bc, lb);
    }
  }
#endif

  // ---------------- Carry combine across the 16 segments -------------------
  sA[w][lane] = Ac;
  sB[w][lane] = Bc;
  __syncthreads();
  if (w == 0) {
    float h = 0.0f;                            // initial state: c[0] = b[0]
#pragma unroll
    for (int j = 0; j < kWaves; ++j) {
      sC[j][lane] = h;                         // carry INTO segment j
      h = fmaf(sA[j][lane], h, sB[j][lane]);
    }
  }
  __syncthreads();

  float h = sC[w][lane];
  float* po = out + ((size_t)bt * kSeq + t0) * kUnits + ucol + lane;

  // ---------------- Phase B: replay segment with carry, stream c out -------
#if HAS_ASYNC_LDS
  issue(0, 0);                                 // re-stage: hits L2 (short reuse)
  issue(kCT, 1);
#pragma unroll
  for (int ck = 0; ck < kChunks; ++ck) {
    if (ck + 1 < kChunks) wait_async_le8(); else wait_async_le0();
    const float* pa = &sTa[w][ck & 1][lane];
    const float* pb = &sTb[w][ck & 1][lane];
    float* o = po + (size_t)(ck * kCT) * kUnits;
#pragma unroll
    for (int tl = 0; tl < kCT; ++tl) {
      const float la = pa[tl * 32];
      const float lb = pb[tl * 32];
      h = fmaf(la, h, lb);
      // Non-temporal: c is write-once; keep L2 for the a/b re-read stream.
      __builtin_nontemporal_store(h, o + (size_t)tl * kUnits);
    }
    if (ck + 2 < kChunks) { wait_ds0(); issue((ck + 2) * kCT, ck & 1); }
  }
#else
  {
    const float* g = base + (size_t)t0 * kRow + lane;
#pragma unroll 4
    for (int t = 0; t < kSeg; ++t) {
      const float la = g[(size_t)t * kRow];
      const float lb = g[(size_t)t * kRow + kUnits];
      h = fmaf(la, h, lb);
      __builtin_nontemporal_store(h, po + (size_t)t * kUnits);
    }
  }
#endif
}

}  // namespace

extern "C" void kernel_launch(void* const* d_in, const int* in_sizes, int n_in,
                              void* d_out, int out_size, void* d_ws, size_t ws_size,
                              hipStream_t stream) {
  (void)in_sizes; (void)n_in; (void)out_size; (void)d_ws; (void)ws_size;
  const float* in = (const float*)d_in[0];   // (16, 2048, 4096) fp32
  float* out = (float*)d_out;                // (16, 2048, 2048) fp32
  dim3 grid(kUnits / 32, kBatch, 1);         // 64 unit-groups x 16 batches
  hipLaunchKernelGGL(tempo_scan_kernel, grid, dim3(kWaves * 32), 0, stream, in, out);
}